// ScaledDotProductAttention_66202625900898
// MI455X (gfx1250) — compile-verified
//
#include <hip/hip_runtime.h>
#include <math.h>

typedef float v2f __attribute__((ext_vector_type(2)));
typedef float v4f __attribute__((ext_vector_type(4)));
typedef float v8f __attribute__((ext_vector_type(8)));

#define NB 16
#define NH 8
#define NL 512
#define ND 64
#define SCALE_F 0.125f
#define BHLD ((long)NB * NH * NL * ND) // 4,194,304

// ---------------------------------------------------------------------------
// Kernel 1: _get_new_rep for K and V.
// grid.x = (B*H*L)/4 = 16384 blocks, grid.y = 2 (0 -> K, 1 -> V), 256 threads.
// Each block handles 4 (b,h,l) positions; thread = (pos-in-block, d).
// ---------------------------------------------------------------------------
__global__ __launch_bounds__(256) void newrep_kernel(
    const float* __restrict__ Ksrc, const float* __restrict__ Vsrc,
    const float* __restrict__ w9, const float* __restrict__ bias1,
    float* __restrict__ Kn, float* __restrict__ Vn)
{
  const float* __restrict__ src = (blockIdx.y == 0) ? Ksrc : Vsrc;
  float* __restrict__ dst       = (blockIdx.y == 0) ? Kn   : Vn;

  __shared__ float red[16][64]; // [pos*4+m][d] partial dot products
  __shared__ float aw[16];      // [pos*4+m] scores -> softmax weights

  const int tid  = threadIdx.x;
  const int posl = tid >> 6;  // 0..3
  const int dd   = tid & 63;
  const int pos  = blockIdx.x * 4 + posl;  // flat (b,h,l)
  const int bb   = pos >> 12;
  const int hh   = (pos >> 9) & 7;
  const int ll   = pos & 511;

  float wv[9];
#pragma unroll
  for (int s = 0; s < 9; ++s) wv[s] = w9[s];
  const float bv = bias1[0];

  // Composed reindexing: (b,h,l,m2,d) -> k5[b, c, m, l3]
  const int c = hh * 64 + (ll >> 3);
  float km[4];
#pragma unroll
  for (int m2 = 0; m2 < 4; ++m2) {
    const int u  = ((ll & 7) << 2) + m2;
    const int m  = u >> 3;
    const int l3 = ((u & 7) << 6) + dd;
    const int base = (bb << 20) + (c << 11) + (m << 9);
    float acc = bv;
#pragma unroll
    for (int s = 0; s < 9; ++s) {
      const int l2 = l3 + s - 8;           // pad of 8 zeros in front of l
      if (l2 >= 0) {
        const int F3 = base + l2;          // flat index into k3
        const int x  = F3 >> 15;
        const int y  = (F3 >> 6) & 511;
        const int i  = (F3 >> 2) & 15;
        const int j  = F3 & 3;
        const int ij = i + j;
        if (ij >= 3) {                     // pad of 3 zeros in front of b axis
          acc += wv[s] * src[x * 8192 + y * 16 + (ij - 3)];
        }
      }
    }
    km[m2] = acc;
  }

  const int tidx = pos * 64 + dd;
  const float tval = src[tidx];

#pragma unroll
  for (int m2 = 0; m2 < 4; ++m2) red[posl * 4 + m2][dd] = tval * km[m2];
  __syncthreads();

  if (tid < 16) {  // reduce over d for each (pos, m)
    float s = 0.f;
    for (int d2 = 0; d2 < 64; ++d2) s += red[tid][d2];
    aw[tid] = s * SCALE_F;
  }
  __syncthreads();
  if (tid < 4) {   // softmax over m=4 for each pos
    const float s0 = aw[tid * 4 + 0], s1 = aw[tid * 4 + 1];
    const float s2 = aw[tid * 4 + 2], s3 = aw[tid * 4 + 3];
    const float mx = fmaxf(fmaxf(s0, s1), fmaxf(s2, s3));
    const float e0 = __expf(s0 - mx), e1 = __expf(s1 - mx);
    const float e2 = __expf(s2 - mx), e3 = __expf(s3 - mx);
    const float inv = 1.f / (e0 + e1 + e2 + e3);
    aw[tid * 4 + 0] = e0 * inv; aw[tid * 4 + 1] = e1 * inv;
    aw[tid * 4 + 2] = e2 * inv; aw[tid * 4 + 3] = e3 * inv;
  }
  __syncthreads();

  float out = tval;  // residual
#pragma unroll
  for (int m2 = 0; m2 < 4; ++m2) out += aw[posl * 4 + m2] * km[m2];
  dst[tidx] = out;
}

// ---------------------------------------------------------------------------
// Kernel 2: attention.  grid = (B*H, L/64), 256 threads = 8 waves.
// Workgroup computes a 64x512 score tile via v_wmma_f32_16x16x4_f32,
// softmaxes rows in LDS (4 threads/row + shfl_xor combine), streams attn
// out with b128 stores, then P @ Vn -> ctx.
// Dynamic LDS: 64*512 floats = 128 KB (fits 320 KB/WGP).
// ---------------------------------------------------------------------------
__global__ __launch_bounds__(256) void attn_kernel(
    const float* __restrict__ Q, const float* __restrict__ Kn,
    const float* __restrict__ Vn, float* __restrict__ ctx,
    float* __restrict__ attn)
{
  HIP_DYNAMIC_SHARED(float, P);  // [64][512]

  const int bh   = blockIdx.x;        // 0..127
  const int q0   = blockIdx.y * 64;   // q tile base
  const int tid  = threadIdx.x;
  const int wv   = tid >> 5;          // wave 0..7
  const int lane = tid & 31;
  const int nloc = lane & 15;         // N index within 16
  const int kb   = (lane >> 4) << 1;  // K sub-offset {0,2}
  const int mhi  = (lane >> 4) << 3;  // C/D row offset {0,8}

  const float* __restrict__ Qb = Q  + (long)bh * NL * ND;
  const float* __restrict__ Kb = Kn + (long)bh * NL * ND;
  const float* __restrict__ Vb = Vn + (long)bh * NL * ND;

  const int qw = wv >> 1;  // q sub-band 0..3 (16 rows each)
  const int nh = wv & 1;   // key half: 16 key-tiles each

  // ---- Preload Q fragments for this wave's q band, pre-scaled by 1/sqrt(D).
  v2f qf[16];
  {
    const float* qp = Qb + (long)(q0 + qw * 16 + nloc) * ND + kb;
#pragma unroll
    for (int ds = 0; ds < 16; ++ds) {
      v2f q2 = *(const v2f*)(qp + ds * 4);
      qf[ds].x = q2.x * SCALE_F;
      qf[ds].y = q2.y * SCALE_F;
    }
  }

  // ---- GEMM1: S = (Q*scale) @ Kn^T, 16 key tiles per wave.
  for (int it = 0; it < 16; ++it) {
    const int k0 = (nh * 16 + it) * 16;
    v8f acc = {};
    const float* bp = Kb + (long)(k0 + nloc) * ND + kb;  // B[d, k] = Kn[k, d]
#pragma unroll
    for (int ds = 0; ds < 16; ++ds) {
      v2f b = *(const v2f*)(bp + ds * 4);
      acc = __builtin_amdgcn_wmma_f32_16x16x4_f32(
          false, qf[ds], false, b, (short)0, acc, false, false);
    }
    const int mbase = qw * 16 + mhi;
#pragma unroll
    for (int r = 0; r < 8; ++r) P[(mbase + r) * 512 + k0 + nloc] = acc[r];
  }
  __syncthreads();

  // ---- Row softmax over 512 keys; 4 threads per row, all 256 threads busy.
  {
    const int row = tid >> 2;  // 0..63
    const int prt = tid & 3;   // 128-col slice
    float* rp = P + row * 512 + prt * 128;

    float m0 = -INFINITY, m1 = -INFINITY, m2 = -INFINITY, m3 = -INFINITY;
    for (int k = 0; k < 128; k += 4) {
      m0 = fmaxf(m0, rp[k + 0]); m1 = fmaxf(m1, rp[k + 1]);
      m2 = fmaxf(m2, rp[k + 2]); m3 = fmaxf(m3, rp[k + 3]);
    }
    float mx = fmaxf(fmaxf(m0, m1), fmaxf(m2, m3));
    mx = fmaxf(mx, __shfl_xor(mx, 1, 32));   // combine 4-lane row group
    mx = fmaxf(mx, __shfl_xor(mx, 2, 32));

    float s0 = 0.f, s1 = 0.f, s2 = 0.f, s3 = 0.f;
    for (int k = 0; k < 128; k += 4) {
      float e0 = __expf(rp[k + 0] - mx); rp[k + 0] = e0; s0 += e0;
      float e1 = __expf(rp[k + 1] - mx); rp[k + 1] = e1; s1 += e1;
      float e2 = __expf(rp[k + 2] - mx); rp[k + 2] = e2; s2 += e2;
      float e3 = __expf(rp[k + 3] - mx); rp[k + 3] = e3; s3 += e3;
    }
    float sum = (s0 + s1) + (s2 + s3);
    sum += __shfl_xor(sum, 1, 32);
    sum += __shfl_xor(sum, 2, 32);
    const float inv = 1.f / sum;
    for (int k = 0; k < 128; ++k) rp[k] *= inv;
  }
  __syncthreads();

  // ---- Stream attn tile out: b128, coalesced across 256 threads.
  {
    v4f* ap = (v4f*)(attn + ((long)bh * NL + q0) * NL);
    const v4f* Pp = (const v4f*)P;
    for (int idx = tid; idx < (64 * 512) / 4; idx += 256) ap[idx] = Pp[idx];
  }

  // ---- GEMM2: ctx = P @ Vn. 16 output tiles (4x4), 2 per wave.
#pragma unroll
  for (int t = 0; t < 2; ++t) {
    const int tw = wv * 2 + t;
    const int mq = tw >> 2;  // q sub-band
    const int nd = tw & 3;   // d sub-band
    v8f acc = {};
    const float* arow = P + (mq * 16 + nloc) * 512 + kb;        // A[q, k] from LDS
    const float* bp   = Vb + (long)kb * ND + nd * 16 + nloc;    // B[k, d] = Vn[k, d]
#pragma unroll 8
    for (int kk = 0; kk < 512; kk += 4) {
      v2f a = *(const v2f*)(arow + kk);
      v2f b;
      b.x = bp[(long)kk * ND];
      b.y = bp[(long)(kk + 1) * ND];
      acc = __builtin_amdgcn_wmma_f32_16x16x4_f32(
          false, a, false, b, (short)0, acc, false, false);
    }
    const int mbase = mq * 16 + mhi;
    float* cp = ctx + ((long)bh * NL + q0 + mbase) * ND + nd * 16 + nloc;
#pragma unroll
    for (int r = 0; r < 8; ++r) cp[(long)r * ND] = acc[r];
  }
}

// ---------------------------------------------------------------------------
extern "C" void kernel_launch(void* const* d_in, const int* in_sizes, int n_in,
                              void* d_out, int out_size, void* d_ws, size_t ws_size,
                              hipStream_t stream) {
  (void)in_sizes; (void)n_in; (void)out_size; (void)ws_size;
  const float* Q  = (const float*)d_in[0];
  const float* K  = (const float*)d_in[1];
  const float* V  = (const float*)d_in[2];
  // d_in[3] = attn_mask: unused by the reference.
  const float* w9 = (const float*)d_in[4];
  const float* b1 = (const float*)d_in[5];

  float* ctx  = (float*)d_out;            // (B,H,L,D)
  float* attn = ctx + BHLD;               // (B,H,L,L)
  float* Kn   = (float*)d_ws;             // workspace: Kn then Vn (16.8 MB each)
  float* Vn   = Kn + BHLD;

  dim3 g1((NB * NH * NL) / 4, 2);
  newrep_kernel<<<g1, 256, 0, stream>>>(K, V, w9, b1, Kn, Vn);

  dim3 g2(NB * NH, NL / 64);
  attn_kernel<<<g2, 256, 64 * 512 * sizeof(float), stream>>>(Q, Kn, Vn, ctx, attn);
}